// LogSparseAttention_77979426226675
// MI455X (gfx1250) — compile-verified
//
#include <hip/hip_runtime.h>
#include <hip/hip_bf16.h>
#include <stdint.h>

// LogSparseAttention for MI455X (gfx1250, wave32).
// B=4, S=2048, D=512, H=8, DEPTH=64, BAND=16.
// fp32 WMMA (v_wmma_f32_16x16x4_f32): reference math is fp32, keep it exact.
// LDS operand buffers laid out so every WMMA fragment is k-contiguous ->
// ds_load_b64 / ds_load_2addr_b64, no repacking movs.
// Attention K-window staging uses CDNA5 async global->LDS (ASYNCcnt path).

typedef __attribute__((ext_vector_type(2))) float v2f;
typedef __attribute__((ext_vector_type(8))) float v8f;

#define S_LEN   2048
#define D_DIM   512
#define H_NUM   8
#define DEPTH   64
#define M_ROWS  8192          // B*S
#define KB      32            // K slab per LDS stage

// ---------------------------------------------------------------------------
// GEMM: Y[M x 512] = X[M x 512] @ W[512 x 512] + bias
// head_layout==1 : scatter into [B,H,S,DEPTH]  (Q/K/V head tensors)
// head_layout==0 : row-major [M x 512]          (final output)
// Block: 256 threads = 8 waves; block tile 128(M) x 128(N); wave tile 32 x 64.
// W slab staged TRANSPOSED in LDS (Wt[col][k]) via 4x4 register micro-tiles so
// B-fragments are contiguous b64 LDS loads. Next slab is prefetched into
// registers while the current slab computes (software pipeline).
// ---------------------------------------------------------------------------
__global__ __launch_bounds__(256) void lsa_gemm512_kernel(
    const float* __restrict__ X, const float* __restrict__ W,
    const float* __restrict__ bias, float* __restrict__ Y, int head_layout) {
  __shared__ float Xs[128][36];   // 128 rows x 32 k (pad->36)
  __shared__ float Wt[128][36];   // 128 cols x 32 k (TRANSPOSED, pad->36)

  const int tid  = threadIdx.x;
  const int lane = tid & 31;
  const int wave = tid >> 5;
  const int wm   = wave & 3;      // 0..3 : 32-row group
  const int wn   = wave >> 2;     // 0..1 : 64-col group
  const int row0 = blockIdx.x * 128;
  const int col0 = blockIdx.y * 128;

  // staging coordinates
  const int wkt = tid >> 5;       // 0..7  : k micro-row of W micro-tile
  const int wct = tid & 31;       // 0..31 : col micro-col of W micro-tile

  float4 rX[4];                   // X slab staging regs (4 float4 / thread)
  float4 rW[4];                   // W micro-tile rows   (4 float4 / thread)

  // ---- load slab 0 into registers ----
  #pragma unroll
  for (int it = 0; it < 4; ++it) {
    const int i = tid + 256 * it;
    rX[it] = *(const float4*)&X[(size_t)(row0 + (i >> 3)) * D_DIM + ((i & 7) << 2)];
  }
  #pragma unroll
  for (int j = 0; j < 4; ++j)
    rW[j] = *(const float4*)&W[(size_t)(wkt * 4 + j) * D_DIM + col0 + wct * 4];

  const v8f vzero = {};
  v8f acc[2][4] = {{vzero, vzero, vzero, vzero}, {vzero, vzero, vzero, vzero}};

  const int m     = lane & 15;
  const int n     = lane & 15;
  const int kbase = (lane >> 4) << 1;   // 0 or 2
  const int mhalf = lane >> 4;

  for (int k0 = 0; k0 < D_DIM; k0 += KB) {
    // ---- store staged registers to LDS (X row-major, W transposed) ----
    #pragma unroll
    for (int it = 0; it < 4; ++it) {
      const int i = tid + 256 * it;
      *(float4*)&Xs[i >> 3][(i & 7) << 2] = rX[it];
    }
    *(float4*)&Wt[wct * 4 + 0][wkt * 4] = make_float4(rW[0].x, rW[1].x, rW[2].x, rW[3].x);
    *(float4*)&Wt[wct * 4 + 1][wkt * 4] = make_float4(rW[0].y, rW[1].y, rW[2].y, rW[3].y);
    *(float4*)&Wt[wct * 4 + 2][wkt * 4] = make_float4(rW[0].z, rW[1].z, rW[2].z, rW[3].z);
    *(float4*)&Wt[wct * 4 + 3][wkt * 4] = make_float4(rW[0].w, rW[1].w, rW[2].w, rW[3].w);
    __syncthreads();

    // ---- issue next slab's global loads (latency hidden by compute) ----
    const int k1 = k0 + KB;
    if (k1 < D_DIM) {
      #pragma unroll
      for (int it = 0; it < 4; ++it) {
        const int i = tid + 256 * it;
        rX[it] = *(const float4*)&X[(size_t)(row0 + (i >> 3)) * D_DIM + k1 + ((i & 7) << 2)];
      }
      #pragma unroll
      for (int j = 0; j < 4; ++j)
        rW[j] = *(const float4*)&W[(size_t)(k1 + wkt * 4 + j) * D_DIM + col0 + wct * 4];
      if (k1 + KB < D_DIM)  // warm L2 for the slab after next -> global_prefetch_b8
        __builtin_prefetch(&X[(size_t)(row0 + (tid >> 1)) * D_DIM + k1 + KB], 0, 1);
    }

    // ---- 8 k-steps x (2 A-tiles x 4 B-tiles) = 64 WMMAs ----
    #pragma unroll
    for (int kk = 0; kk < KB; kk += 4) {
      v2f a0, a1;
      a0.x = Xs[wm * 32 + m][kk + kbase];
      a0.y = Xs[wm * 32 + m][kk + kbase + 1];
      a1.x = Xs[wm * 32 + 16 + m][kk + kbase];
      a1.y = Xs[wm * 32 + 16 + m][kk + kbase + 1];
      #pragma unroll
      for (int nt = 0; nt < 4; ++nt) {
        v2f b;
        b.x = Wt[wn * 64 + nt * 16 + n][kk + kbase];      // contiguous -> b64
        b.y = Wt[wn * 64 + nt * 16 + n][kk + kbase + 1];
        acc[0][nt] = __builtin_amdgcn_wmma_f32_16x16x4_f32(
            false, a0, false, b, (short)0, acc[0][nt], false, false);
        acc[1][nt] = __builtin_amdgcn_wmma_f32_16x16x4_f32(
            false, a1, false, b, (short)0, acc[1][nt], false, false);
      }
    }
    __syncthreads();
  }

  // ---- store with fused bias. C layout: VGPR v -> row v+8*mhalf, col lane&15.
  #pragma unroll
  for (int t = 0; t < 2; ++t) {
    #pragma unroll
    for (int nt = 0; nt < 4; ++nt) {
      const int col = col0 + wn * 64 + nt * 16 + n;
      const float bv = bias[col];
      #pragma unroll
      for (int v = 0; v < 8; ++v) {
        const int grow = row0 + wm * 32 + t * 16 + mhalf * 8 + v;
        const float val = acc[t][nt][v] + bv;
        if (head_layout) {
          const int b = grow >> 11;          // / S_LEN
          const int s = grow & (S_LEN - 1);
          const int h = col >> 6;            // / DEPTH
          const int d = col & (DEPTH - 1);
          Y[((size_t)(b * H_NUM + h) * S_LEN + s) * DEPTH + d] = val;
        } else {
          Y[(size_t)grow * D_DIM + col] = val;
        }
      }
    }
  }
}

// ---------------------------------------------------------------------------
// Banded attention: one wave per (b,h, 16-query) tile.
// Key window [q0-16, q0+31] (48 keys) covers every |q-k|<=16 pair.
// logits = Q(16x64) @ K^T(64x48) -> masked softmax (*0.125) -> @ V(48x64).
// K staged via CDNA5 async global->LDS copies (no VGPR round-trip, ASYNCcnt);
// V staged TRANSPOSED (Vt[depth][key]) so P@V B-frags are contiguous b64.
// Block: 64 threads = 2 waves with private LDS slices (no barriers needed).
// ---------------------------------------------------------------------------
__global__ __launch_bounds__(64) void lsa_band_attn_kernel(
    const float* __restrict__ Qh, const float* __restrict__ Kh,
    const float* __restrict__ Vh, float* __restrict__ O) {
  __shared__ float Ks[2][48][68];   // [key][depth]
  __shared__ float Vt[2][64][52];   // [depth][key]  (transposed)
  __shared__ float Ps[2][16][52];   // probabilities [query][key]

  const int lane = threadIdx.x & 31;
  const int wave = threadIdx.x >> 5;
  const int tile = blockIdx.x * 2 + wave;          // 0 .. 4095
  const int qt   = tile & 127;                     // S/16 tiles
  const int bh   = tile >> 7;                      // 0 .. 31
  const int q0   = qt * 16;
  const int k0   = q0 - 16;

  const float* Qb = Qh + (size_t)bh * S_LEN * DEPTH;
  const float* Kb = Kh + (size_t)bh * S_LEN * DEPTH;
  const float* Vb = Vh + (size_t)bh * S_LEN * DEPTH;

  // ---- stage K via async global->LDS b128 (in-range) / zero-fill (edges) --
  for (int i = lane; i < 48 * 16; i += 32) {
    const int r  = i >> 4;
    const int c4 = (i & 15) << 2;
    const int kg = k0 + r;
    if (kg >= 0 && kg < S_LEN) {
      const uint32_t dsaddr = (uint32_t)(uintptr_t)&Ks[wave][r][c4];
      const float* src = &Kb[(size_t)kg * DEPTH + c4];
      asm volatile("global_load_async_to_lds_b128 %0, %1, off"
                   :: "v"(dsaddr), "v"(src) : "memory");
    } else {
      *(float4*)&Ks[wave][r][c4] = make_float4(0.f, 0.f, 0.f, 0.f);
    }
  }
  // ---- stage V transposed via 4x4 micro-tiles: 12(k) x 16(d) tiles, 6/lane -
  for (int mt = lane; mt < 12 * 16; mt += 32) {
    const int kt = mt >> 4;      // key micro-row  (0..11)
    const int ct = mt & 15;      // depth micro-col (0..15)
    float4 r[4];
    #pragma unroll
    for (int j = 0; j < 4; ++j) {
      const int kg = k0 + kt * 4 + j;
      r[j] = make_float4(0.f, 0.f, 0.f, 0.f);
      if (kg >= 0 && kg < S_LEN) r[j] = *(const float4*)&Vb[(size_t)kg * DEPTH + ct * 4];
    }
    *(float4*)&Vt[wave][ct * 4 + 0][kt * 4] = make_float4(r[0].x, r[1].x, r[2].x, r[3].x);
    *(float4*)&Vt[wave][ct * 4 + 1][kt * 4] = make_float4(r[0].y, r[1].y, r[2].y, r[3].y);
    *(float4*)&Vt[wave][ct * 4 + 2][kt * 4] = make_float4(r[0].z, r[1].z, r[2].z, r[3].z);
    *(float4*)&Vt[wave][ct * 4 + 3][kt * 4] = make_float4(r[0].w, r[1].w, r[2].w, r[3].w);
  }
  // all async K tiles landed in this wave's LDS slice
  asm volatile("s_wait_asynccnt 0x0" ::: "memory");

  const int m     = lane & 15;
  const int n     = lane & 15;
  const int kbase = (lane >> 4) << 1;
  const int mhalf = lane >> 4;

  // ---- logits = Q @ K^T : 16 k-steps x 3 key tiles = 48 WMMAs ----
  const v8f vzero = {};
  v8f lg[3] = {vzero, vzero, vzero};
  const float* qrow = Qb + (size_t)(q0 + m) * DEPTH;
  #pragma unroll
  for (int kk = 0; kk < DEPTH; kk += 4) {
    v2f a;
    const float2 qv = *(const float2*)&qrow[kk + kbase];
    a.x = qv.x; a.y = qv.y;
    #pragma unroll
    for (int nt = 0; nt < 3; ++nt) {
      v2f b;                                   // B[kdim][key] = K[key][kdim]
      b.x = Ks[wave][nt * 16 + n][kk + kbase];
      b.y = Ks[wave][nt * 16 + n][kk + kbase + 1];
      lg[nt] = __builtin_amdgcn_wmma_f32_16x16x4_f32(
          false, a, false, b, (short)0, lg[nt], false, false);
    }
  }
  // spill logits for the row softmax
  #pragma unroll
  for (int nt = 0; nt < 3; ++nt)
    #pragma unroll
    for (int v = 0; v < 8; ++v)
      Ps[wave][mhalf * 8 + v][nt * 16 + n] = lg[nt][v];

  // ---- masked softmax over the 48-key window (reference math exactly) ----
  if (lane < 16) {
    const int qg = q0 + m;
    const float scale = 0.125f;              // 1/sqrt(64)
    float xs[48];
    float mx = -3.0e38f;
    #pragma unroll
    for (int j = 0; j < 48; ++j) {
      const int kg = k0 + j;
      const int dd = qg - kg;
      const bool keep = (kg >= 0) & (kg < S_LEN) & (dd >= -16) & (dd <= 16);
      const float x = keep ? Ps[wave][m][j] * scale : -1.0e9f;
      xs[j] = x;
      mx = fmaxf(mx, x);
    }
    float sum = 0.f;
    #pragma unroll
    for (int j = 0; j < 48; ++j) {
      const float e = __expf(xs[j] - mx);
      xs[j] = e;
      sum += e;
    }
    const float inv = 1.f / sum;
    #pragma unroll
    for (int j = 0; j < 48; ++j) Ps[wave][m][j] = xs[j] * inv;
  }

  // ---- out = P(16x48) @ V(48x64) : 12 k-steps x 4 depth tiles = 48 WMMAs --
  v8f out[4] = {vzero, vzero, vzero, vzero};
  #pragma unroll
  for (int kk = 0; kk < 48; kk += 4) {
    v2f a;
    a.x = Ps[wave][m][kk + kbase];
    a.y = Ps[wave][m][kk + kbase + 1];
    #pragma unroll
    for (int nt = 0; nt < 4; ++nt) {
      v2f b;                                   // B[key][depth] = Vt[depth][key]
      b.x = Vt[wave][nt * 16 + n][kk + kbase];      // contiguous -> b64
      b.y = Vt[wave][nt * 16 + n][kk + kbase + 1];
      out[nt] = __builtin_amdgcn_wmma_f32_16x16x4_f32(
          false, a, false, b, (short)0, out[nt], false, false);
    }
  }

  // ---- store to merged-head [B*S, 512]: row = b*S + s, col = h*64 + d ----
  const int b = bh >> 3;
  const int h = bh & 7;
  #pragma unroll
  for (int nt = 0; nt < 4; ++nt) {
    const int col = h * DEPTH + nt * 16 + n;
    #pragma unroll
    for (int v = 0; v < 8; ++v) {
      const int s = q0 + mhalf * 8 + v;
      O[(size_t)(b * S_LEN + s) * D_DIM + col] = out[nt][v];
    }
  }
}

// ---------------------------------------------------------------------------
extern "C" void kernel_launch(void* const* d_in, const int* in_sizes, int n_in,
                              void* d_out, int out_size, void* d_ws, size_t ws_size,
                              hipStream_t stream) {
  const float* q  = (const float*)d_in[0];
  const float* k  = (const float*)d_in[1];
  const float* v  = (const float*)d_in[2];
  const float* wq = (const float*)d_in[3];
  const float* bq = (const float*)d_in[4];
  const float* wk = (const float*)d_in[5];
  const float* bk = (const float*)d_in[6];
  const float* wv = (const float*)d_in[7];
  const float* bv = (const float*)d_in[8];
  const float* wo = (const float*)d_in[9];
  const float* bo = (const float*)d_in[10];

  const size_t TENS = (size_t)M_ROWS * D_DIM;   // 4,194,304 floats = 16 MB
  float* qh = (float*)d_ws;
  float* kh = qh + TENS;
  float* vh = kh + TENS;
  float* ao = vh + TENS;

  dim3 ggrid(M_ROWS / 128, D_DIM / 128);        // (64, 4)
  dim3 gblock(256);

  // Q/K/V projections into [B,H,S,64] head layout.
  lsa_gemm512_kernel<<<ggrid, gblock, 0, stream>>>(q, wq, bq, qh, 1);
  lsa_gemm512_kernel<<<ggrid, gblock, 0, stream>>>(k, wk, bk, kh, 1);
  lsa_gemm512_kernel<<<ggrid, gblock, 0, stream>>>(v, wv, bv, vh, 1);

  // Banded attention: 4096 query tiles, 2 waves per block.
  lsa_band_attn_kernel<<<2048, 64, 0, stream>>>(qh, kh, vh, ao);

  // Output projection into d_out [B,S,D].
  lsa_gemm512_kernel<<<ggrid, gblock, 0, stream>>>(ao, wo, bo, (float*)d_out, 0);
}